// HeteroDotProductPredictor_13503377179006
// MI455X (gfx1250) — compile-verified
//
#include <hip/hip_runtime.h>
#include <hip/hip_bf16.h>
#include <stdint.h>

typedef __attribute__((ext_vector_type(2))) float v2f;
typedef __attribute__((ext_vector_type(8))) float v8f;

// Branchless diag extraction: select c[ci] with a cndmask tree (no scratch, no branches).
__device__ __forceinline__ float sel8(const v8f c, unsigned ci) {
    float v0 = (ci & 1u) ? c[1] : c[0];
    float v1 = (ci & 1u) ? c[3] : c[2];
    float v2 = (ci & 1u) ? c[5] : c[4];
    float v3 = (ci & 1u) ? c[7] : c[6];
    float w0 = (ci & 2u) ? v1 : v0;
    float w1 = (ci & 2u) ? v3 : v2;
    return (ci & 4u) ? w1 : w0;
}

// One wave computes 32 edge dot-products as TWO independent 16-edge tiles via
// V_WMMA_F32_16X16X4_F32 (fp32-exact), giving two independent accumulator
// chains to hide WMMA latency:
//   Dt(16x16) += A(16x4: rows = h[src[tile]]) * B(4x16: cols = h[dst[tile]])
// diag(Dt) = the 16 dot products of tile t.
template <int DV>
__global__ __launch_bounds__(256) void edge_dot_wmma_f32(
    const float* __restrict__ h,
    const float* __restrict__ bvec,
    const long long* __restrict__ src,
    const long long* __restrict__ dst,
    float* __restrict__ out,
    int E, int Drt)
{
    const int D = (DV > 0) ? DV : Drt;

    const unsigned lane = threadIdx.x & 31u;
    const unsigned wv   = threadIdx.x >> 5;
    const int wave = blockIdx.x * (int)(blockDim.x >> 5) + (int)wv;
    const int e0   = wave * 32;                 // 2 tiles of 16 edges
    if (e0 >= E) return;                        // wave-uniform; EXEC stays all-1s

    // ---- operand row fetch (lane L serves edge-slot L&15 of each tile) ----
    const int slot = (int)(lane & 15u);
    int eA = e0 + slot;        if (eA > E - 1) eA = E - 1;
    int eB = e0 + 16 + slot;   if (eB > E - 1) eB = E - 1;

    const long long s0 = src[eA], d0 = dst[eA];
    const long long s1 = src[eB], d1 = dst[eB];
    const int kOff = (int)((lane >> 4) << 1);   // 0 or 2: which K-pair of the chunk

    const float* pu0 = h + (size_t)s0 * (size_t)D + kOff;
    const float* pv0 = h + (size_t)d0 * (size_t)D + kOff;
    const float* pu1 = h + (size_t)s1 * (size_t)D + kOff;
    const float* pv1 = h + (size_t)d1 * (size_t)D + kOff;

    // ---- hoist the store-phase b-gathers so their latency hides under the
    //      WMMA loop. Writer mapping (C/D layout): diag (j,j) lives at
    //      lane j, VGPR j (j<8) or lane j+8, VGPR j-8 (j>=8).
    const bool active = (lane < 8u) | ((lane >= 16u) & (lane < 24u));
    const unsigned ci = (lane < 8u) ? lane : (lane - 16u);        // VGPR index
    const int      j  = (lane < 8u) ? (int)lane : (int)lane - 8;  // edge-in-tile
    int ew0 = e0 + j;        if (ew0 > E - 1) ew0 = E - 1;
    int ew1 = e0 + 16 + j;   if (ew1 > E - 1) ew1 = E - 1;
    const float bsum0 = bvec[src[ew0]] + bvec[dst[ew0]];
    const float bsum1 = bvec[src[ew1]] + bvec[dst[ew1]];

    // ---- two independent WMMA accumulate chains over the K dimension ----
    v8f c0 = {};
    v8f c1 = {};
    #pragma unroll 8
    for (int k0 = 0; k0 < D; k0 += 4) {
        v2f a0 = *(const v2f*)(pu0 + k0);
        v2f b0 = *(const v2f*)(pv0 + k0);
        v2f a1 = *(const v2f*)(pu1 + k0);
        v2f b1 = *(const v2f*)(pv1 + k0);
        // 8 args: (neg_a, A, neg_b, B, c_mod, C, reuse_a, reuse_b)
        c0 = __builtin_amdgcn_wmma_f32_16x16x4_f32(
                 false, a0, false, b0, (short)0, c0, false, false);
        c1 = __builtin_amdgcn_wmma_f32_16x16x4_f32(
                 false, a1, false, b1, (short)0, c1, false, false);
    }

    // ---- branchless diagonal extraction + predicated store ----
    const float v0 = sel8(c0, ci) + bsum0;
    const float v1 = sel8(c1, ci) + bsum1;
    if (active && (e0 + j)      < E) out[e0 + j]      = v0;
    if (active && (e0 + 16 + j) < E) out[e0 + 16 + j] = v1;
}

extern "C" void kernel_launch(void* const* d_in, const int* in_sizes, int n_in,
                              void* d_out, int out_size, void* d_ws, size_t ws_size,
                              hipStream_t stream) {
    const float*     h   = (const float*)d_in[0];      // [N, D] f32
    const float*     b   = (const float*)d_in[1];      // [N, 1] f32
    const long long* src = (const long long*)d_in[2];  // [E] int64
    const long long* dst = (const long long*)d_in[3];  // [E] int64
    float*           out = (float*)d_out;              // [E, 1] f32

    const int E = in_sizes[2];
    const int N = in_sizes[1];
    const int D = (N > 0) ? (in_sizes[0] / N) : 128;   // 128 in reference

    const int threads       = 256;                     // 8 waves (wave32)
    const int edgesPerBlock = (threads / 32) * 32;     // 256 edges / block
    const int grid          = (E + edgesPerBlock - 1) / edgesPerBlock;

    if (D == 128) {
        edge_dot_wmma_f32<128><<<grid, threads, 0, stream>>>(h, b, src, dst, out, E, D);
    } else {
        edge_dot_wmma_f32<0><<<grid, threads, 0, stream>>>(h, b, src, dst, out, E, D);
    }
}